// NONLocalBlock2D_13692355739916
// MI455X (gfx1250) — compile-verified
//
#include <hip/hip_runtime.h>

// Batched GEMV: y[b,n] = sum_m W[b,n,m] * x[b,m];  B=3, N=8704.
// HBM-bound (0.5 FLOP/byte, 909 MB weight stream -> ~39us floor @ 23.3 TB/s).
// V_WMMA_F32_16X16X4_F32 is the MAC/reduction engine; TDM stages x into LDS.

typedef float v2f __attribute__((ext_vector_type(2)));
typedef float v8f __attribute__((ext_vector_type(8)));
typedef unsigned int u32x4 __attribute__((ext_vector_type(4)));
typedef int i32x4 __attribute__((ext_vector_type(4)));
typedef int i32x8 __attribute__((ext_vector_type(8)));

#define BATCH   3
#define NDIM    8704
#define TILE    16
#define TPB     256
#define NWAVE   (TPB / 32)            // 8 waves per block (wave32)
#define KSLICE  (NDIM / NWAVE)        // 1088 columns per wave
#define KSTEP4  (KSLICE / 4)          // 272 wmma steps of K=4
#define TILES_PER_BATCH (NDIM / TILE) // 544

__global__ __launch_bounds__(TPB)
void nonlocal_gemv_wmma(const float* __restrict__ W,
                        const float* __restrict__ x,
                        float* __restrict__ y,
                        int out_size) {
    __shared__ float sx[NDIM];        // 34,816 B: full x[b] staged in LDS
    __shared__ float yred[TILE];      // cross-wave partial-sum reduction

    const int tid  = threadIdx.x;
    const int wave = tid >> 5;
    const int lane = tid & 31;
    const int b    = blockIdx.x / TILES_PER_BATCH;
    const int tile = blockIdx.x % TILES_PER_BATCH;

    const float* xb = x + (size_t)b * NDIM;

#if __has_builtin(__builtin_amdgcn_tensor_load_to_lds)
    // --- Stage x[b] via the Tensor Data Mover (wave 0 issues one 1-D DMA) ---
    if (wave == 0) {
        const unsigned lds_off = (unsigned)(uintptr_t)&sx[0];     // low 32 bits = LDS byte offset
        const unsigned long long ga = (unsigned long long)(uintptr_t)xb;
        // D# group 0: count=1 | lds_addr | global_addr[56:0] | type=2 ("image")
        u32x4 g0 = { 1u,
                     lds_off,
                     (unsigned)(ga & 0xFFFFFFFFu),
                     (unsigned)((ga >> 32) & 0x01FFFFFFu) | (2u << 30) };
        // D# group 1: data_size=4B; tensor_dim0=tile_dim0=8704 (1-D tile),
        // tensor_dim1=1, tile_dim1/2=0 (unused), tensor_dim0_stride=8704.
        i32x8 g1 = { (int)(2u << 16),                         // data_size=2 -> 4 bytes
                     (int)((NDIM & 0xFFFF) << 16),            // tensor_dim0[15:0]
                     (int)((NDIM >> 16) | (1u << 16)),        // tensor_dim0[31:16] | tensor_dim1=1
                     (int)((NDIM & 0xFFFF) << 16),            // tile_dim0 = 8704
                     0,                                       // tile_dim1=0, tile_dim2=0
                     (int)NDIM,                               // tensor_dim0_stride[31:0]
                     0, 0 };
        i32x4 gz = { 0, 0, 0, 0 };                            // groups 2/3 unused (<=2D)
#if __clang_major__ >= 23
        i32x8 g4 = { 0, 0, 0, 0, 0, 0, 0, 0 };
        __builtin_amdgcn_tensor_load_to_lds(g0, g1, gz, gz, g4, 0);
#else
        __builtin_amdgcn_tensor_load_to_lds(g0, g1, gz, gz, 0);
#endif
        __builtin_amdgcn_s_wait_tensorcnt(0);                 // s_wait_tensorcnt 0
    }
#else
    for (int i = tid; i < NDIM; i += TPB) sx[i] = xb[i];
#endif
    if (tid < TILE) yred[tid] = 0.0f;
    __syncthreads();                  // releases waves 1..7 after TDM completes

    // A-fragment addressing per ISA 16x4 f32 layout:
    //   lanes 0-15 : row = lane,    K = k+0, k+1   (VGPR0, VGPR1)
    //   lanes 16-31: row = lane-16, K = k+2, k+3
    const int row   = tile * TILE + (lane & 15);
    const int khalf = (lane >> 4) * 2;                 // 0 or 2
    const float* wrow = W + (size_t)b * NDIM * NDIM + (size_t)row * NDIM;
    const bool  bsel  = ((lane & 15) == 0);            // lanes 0 and 16 carry B col 0

    v8f c = {};                       // D accumulator, column 0 = partial y
    const int k0 = wave * KSLICE;

    for (int so = 0; so < KSTEP4; so += 8) {
        // Speculative prefetch of this lane's weight stream ~2 KB ahead.
        __builtin_prefetch(wrow + k0 + (so + 8) * 4 + khalf + 512, 0, 1);
        #pragma unroll
        for (int si = 0; si < 8; ++si) {
            const int k = k0 + (so + si) * 4;
            // 8B per lane; 16 row-streams per wave, lanes 0-15/16-31 cover
            // alternating 8B halves of each 16B chunk of a row.
            v2f a = *(const v2f*)(wrow + k + khalf);
            // One ds_load_b64 for all lanes: khalf makes lanes<16 read
            // x[k],x[k+1] and lanes>=16 read x[k+2],x[k+3]; mask keeps only
            // lanes 0 and 16 (B-matrix column N=0), rest of B is zero.
            v2f xv = *(const v2f*)(sx + k + khalf);
            v2f bm;
            bm.x = bsel ? xv.x : 0.0f;
            bm.y = bsel ? xv.y : 0.0f;
            c = __builtin_amdgcn_wmma_f32_16x16x4_f32(
                    /*neg_a=*/false, a, /*neg_b=*/false, bm,
                    /*c_mod=*/(short)0, c, /*reuse_a=*/false, /*reuse_b=*/false);
        }
    }

    // D column 0: lane 0 holds rows 0..7 (VGPR 0..7), lane 16 rows 8..15.
    if (lane == 0) {
        #pragma unroll
        for (int r = 0; r < 8; ++r) atomicAdd(&yred[r], c[r]);     // ds_add_f32
    }
    if (lane == 16) {
        #pragma unroll
        for (int r = 0; r < 8; ++r) atomicAdd(&yred[8 + r], c[r]);
    }
    __syncthreads();

    if (tid < TILE) {
        const long idx = (long)b * NDIM + (long)tile * TILE + tid;
        if (idx < (long)out_size) y[idx] = yred[tid];
    }
}

extern "C" void kernel_launch(void* const* d_in, const int* in_sizes, int n_in,
                              void* d_out, int out_size, void* d_ws, size_t ws_size,
                              hipStream_t stream) {
    const float* source = (const float*)d_in[0];  // (B, C, H, W) -> x
    const float* weight = (const float*)d_in[1];  // (B, N, N)
    float* out = (float*)d_out;

    const int grid = BATCH * TILES_PER_BATCH;     // 1632 blocks
    nonlocal_gemv_wmma<<<grid, TPB, 0, stream>>>(weight, source, out, out_size);
}